// LocalSSIMLoss_32581621907915
// MI455X (gfx1250) — compile-verified
//
#include <hip/hip_runtime.h>

// ---------------------------------------------------------------------------
// SSIM loss on MI455X (gfx1250): separable 11-tap Gaussian conv as banded
// matrix products on V_WMMA_F32_16X16X4_F32 (fp32-exact). All 5 local stats
// fused into one k-sweep so region LDS loads and banded-weight fragments are
// shared across 5 WMMAs. Weight lookups use a zero-padded table (no branches).
// ---------------------------------------------------------------------------

typedef float v2f __attribute__((ext_vector_type(2)));
typedef float v8f __attribute__((ext_vector_type(8)));

#define IMG_H 512
#define IMG_W 512
#define N_PLANES 48            // B*C = 16*3
#define ROWS_R 32              // 26 valid halo rows + slack (rows 26..31 weight-zeroed)
#define COLS_R 76              // 74 valid halo cols + slack (K padded to 28)
#define HS_H   34              // H-scratch column stride (rows 0..31, bank-friendly, 8B-align)
#define HS_PER_STAT (16 * HS_H)
#define C1_SSIM 1.0e-4f        // 0.01^2
#define C2_SSIM 9.0e-4f        // 0.03^2

#define WMMA_F32(a, b, c) \
    __builtin_amdgcn_wmma_f32_16x16x4_f32(false, (a), false, (b), (short)0, (c), false, false)

__global__ __launch_bounds__(128) void ssim_tile_kernel(
    const float* __restrict__ inp,   // d_in[0]
    const float* __restrict__ tgt,   // d_in[1]
    float* __restrict__ partials)    // one float per block
{
    __shared__ float Treg[ROWS_R * COLS_R];       // 9728 B
    __shared__ float Ireg[ROWS_R * COLS_R];       // 9728 B
    __shared__ float Hs[4 * 5 * HS_PER_STAT];     // 43520 B (4 waves x 5 stats, col-major)
    __shared__ float wpad[64];                    // wpad[16+d] = w[d] for d in [0,10], else 0
    __shared__ float wave_sums[4];

    const int tid  = threadIdx.x;
    const int lane = tid & 31;
    const int wv   = tid >> 5;            // wave id 0..3
    const int half = lane >> 4;           // 0: lanes 0-15, 1: lanes 16-31
    const int ln   = lane & 15;
    const int tx   = wv;                  // wave tile col within block (0..3)

    const int bid = blockIdx.x;
    const int img = bid >> 8;             // 256 blocks per plane (32 row x 8 col)
    const int rem = bid & 255;
    const int by0 = (rem >> 3) * 16;      // 16 output rows / block
    const int bx0 = (rem & 7) * 64;       // 64 output cols / block

    const float* __restrict__ tgt_p = tgt + (size_t)img * (IMG_H * IMG_W);
    const float* __restrict__ inp_p = inp + (size_t)img * (IMG_H * IMG_W);

    // --- zero-padded normalized gaussian taps (exp(-x^2/4.5), sum-normalized)
    if (tid < 64) {
        int d = tid - 16;
        float val = 0.0f;
        if ((unsigned)d < 11u) {
            float s = 0.0f;
            for (int t = 0; t < 11; ++t) {
                float x = (float)(t - 5);
                s += __expf(-(x * x) / 4.5f);
            }
            float x = (float)(d - 5);
            val = __expf(-(x * x) / 4.5f) / s;
        }
        wpad[tid] = val;
    }

    // --- cooperative halo load (zero padding outside the image) -------------
    for (int idx = tid; idx < ROWS_R * COLS_R; idx += 128) {
        int row = idx / COLS_R;
        int col = idx - row * COLS_R;
        int gy = by0 - 5 + row;
        int gx = bx0 - 5 + col;
        float tv = 0.0f, iv = 0.0f;
        if ((unsigned)gy < IMG_H && (unsigned)gx < IMG_W) {
            int off = gy * IMG_W + gx;
            tv = tgt_p[off];
            iv = inp_p[off];
        }
        Treg[idx] = tv;
        Ireg[idx] = iv;
    }
    __syncthreads();

    // --- banded weight fragments: serve as horizontal B and vertical A ------
    // W[c][n] = w[c-n] if 0 <= c-n <= 10, else 0 (exact zeros kill halo slack)
    v2f wf[7];
#pragma unroll
    for (int k = 0; k < 7; ++k) {
        int id = 16 + (4 * k + 2 * half) - ln;    // in [1, 43]
        wf[k].x = wpad[id];
        wf[k].y = wpad[id + 1];
    }

    float* Hw = &Hs[wv * 5 * HS_PER_STAT];

    // ------------------------------------------------------------------------
    // Horizontal: H_s(32x16) = Region_s(32x28) x Wh(28x16), 2 row-tiles x 7 k.
    // A layout (16x4 f32): lane<16 -> {K,K+1}, lane>=16 -> {K+2,K+3}, M=ln.
    // All 5 stats share the region loads and the B fragment.
    // ------------------------------------------------------------------------
#pragma unroll
    for (int rt = 0; rt < 2; ++rt) {
        const int rb = rt * 16;
        const int r  = rb + ln;                         // region row (M)
        const float* __restrict__ Trow = &Treg[r * COLS_R + tx * 16];
        const float* __restrict__ Irow = &Ireg[r * COLS_R + tx * 16];

        v8f h[5];
#pragma unroll
        for (int s = 0; s < 5; ++s) h[s] = (v8f){0.f, 0.f, 0.f, 0.f, 0.f, 0.f, 0.f, 0.f};

#pragma unroll
        for (int k = 0; k < 7; ++k) {
            const int kb = 4 * k + 2 * half;            // local K base
            float t0 = Trow[kb], t1 = Trow[kb + 1];     // ds_load_b64
            float i0 = Irow[kb], i1 = Irow[kb + 1];     // ds_load_b64
            v2f b = wf[k];
            v2f a0 = {t0, t1};
            v2f a1 = {i0, i1};
            v2f a2 = {t0 * t0, t1 * t1};
            v2f a3 = {i0 * i0, i1 * i1};
            v2f a4 = {t0 * i0, t1 * i1};
            h[0] = WMMA_F32(a0, b, h[0]);
            h[1] = WMMA_F32(a1, b, h[1]);
            h[2] = WMMA_F32(a2, b, h[2]);
            h[3] = WMMA_F32(a3, b, h[3]);
            h[4] = WMMA_F32(a4, b, h[4]);
        }

        // store H column-major: column ln, rows rb+8*half+j (contiguous b64s)
        const int base = ln * HS_H + rb + 8 * half;
#pragma unroll
        for (int s = 0; s < 5; ++s) {
            float* dst = Hw + s * HS_PER_STAT + base;
#pragma unroll
            for (int p = 0; p < 4; ++p) {
                v2f pr = {h[s][2 * p], h[s][2 * p + 1]};
                *(v2f*)(dst + 2 * p) = pr;              // ds_store_b64
            }
        }
    }

    // ------------------------------------------------------------------------
    // Vertical: Out_s(16x16) = Wv(16x28) x H_s(28x16), 7 k-steps.
    // A = wf[k] (same banded matrix); B read column-major (contiguous b64).
    // ------------------------------------------------------------------------
    v8f st[5];
#pragma unroll
    for (int s = 0; s < 5; ++s) st[s] = (v8f){0.f, 0.f, 0.f, 0.f, 0.f, 0.f, 0.f, 0.f};

#pragma unroll
    for (int k = 0; k < 7; ++k) {
        const int kb = 4 * k + 2 * half;                // K = H row
        v2f a = wf[k];
#pragma unroll
        for (int s = 0; s < 5; ++s) {
            const float* src = Hw + s * HS_PER_STAT + ln * HS_H + kb;
            v2f b = *(const v2f*)src;                   // ds_load_b64
            st[s] = WMMA_F32(a, b, st[s]);
        }
    }

    // ------------------- elementwise SSIM + masked sum ----------------------
    float partial = 0.0f;
#pragma unroll
    for (int j = 0; j < 8; ++j) {
        const int m = j + 8 * half;                     // tile row
        float mu1 = st[0][j];
        float mu2 = st[1][j];
        float e11 = st[2][j];
        float e22 = st[3][j];
        float e12 = st[4][j];
        float mu1_sq  = mu1 * mu1;
        float mu2_sq  = mu2 * mu2;
        float mu1_mu2 = mu1 * mu2;
        float sig1 = e11 - mu1_sq;
        float sig2 = e22 - mu2_sq;
        float s12  = e12 - mu1_mu2;
        float num = (2.0f * mu1_mu2 + C1_SSIM) * (2.0f * s12 + C2_SSIM);
        float den = (mu1_sq + mu2_sq + C1_SSIM) * (sig1 + sig2 + C2_SSIM);
        float ssim = num / den;
        float tval = Treg[(m + 5) * COLS_R + (tx * 16 + ln + 5)];
        float mask = (tval > 0.0f) ? 1.0f : 0.0f;
        partial += mask * (1.0f - ssim);
    }

    // wave reduce (wave32)
#pragma unroll
    for (int off = 16; off >= 1; off >>= 1)
        partial += __shfl_xor(partial, off, 32);
    if (lane == 0) wave_sums[wv] = partial;
    __syncthreads();
    if (tid == 0) {
        float s = wave_sums[0] + wave_sums[1] + wave_sums[2] + wave_sums[3];
        partials[bid] = s;
    }
}

__global__ __launch_bounds__(256) void ssim_reduce_kernel(
    const float* __restrict__ partials, float* __restrict__ out,
    int n, float scale)
{
    __shared__ float sm[256];
    float s = 0.0f;
    for (int i = threadIdx.x; i < n; i += 256) s += partials[i];
    sm[threadIdx.x] = s;
    __syncthreads();
    for (int off = 128; off > 0; off >>= 1) {
        if (threadIdx.x < off) sm[threadIdx.x] += sm[threadIdx.x + off];
        __syncthreads();
    }
    if (threadIdx.x == 0) out[0] = sm[0] * scale;
}

extern "C" void kernel_launch(void* const* d_in, const int* in_sizes, int n_in,
                              void* d_out, int out_size, void* d_ws, size_t ws_size,
                              hipStream_t stream) {
    const float* inp = (const float*)d_in[0];   // "input"
    const float* tgt = (const float*)d_in[1];   // "target"
    float* out = (float*)d_out;
    float* partials = (float*)d_ws;             // needs 12288 floats = 48 KB

    const int nblocks = N_PLANES * (IMG_H / 16) * (IMG_W / 64);   // 12288
    const float scale = 1.0f / (float)(N_PLANES * IMG_H * IMG_W);

    ssim_tile_kernel<<<nblocks, 128, 0, stream>>>(inp, tgt, partials);
    ssim_reduce_kernel<<<1, 256, 0, stream>>>(partials, out, nblocks, scale);
}